// GCNFeatureExtractor_12910671691963
// MI455X (gfx1250) — compile-verified
//
#include <hip/hip_runtime.h>

typedef __attribute__((ext_vector_type(2))) float v2f;
typedef __attribute__((ext_vector_type(8))) float v8f;

// ---------------------------------------------------------------------------
// small utility kernels
// ---------------------------------------------------------------------------
__global__ void fill_f32(float* __restrict__ p, float v, int n) {
    int i = blockIdx.x * blockDim.x + threadIdx.x;
    if (i < n) p[i] = v;
}

__global__ void deg_accum(const int* __restrict__ dst, float* __restrict__ deg, int E) {
    int e = blockIdx.x * blockDim.x + threadIdx.x;
    if (e < E) atomicAdd(&deg[dst[e]], 1.0f);
}

__global__ void deg_to_dinv(float* __restrict__ d, int n) {
    int i = blockIdx.x * blockDim.x + threadIdx.x;
    if (i < n) d[i] = 1.0f / sqrtf(d[i]);   // deg >= 1 always (self loop)
}

// ---------------------------------------------------------------------------
// f32 WMMA GEMM:  D[M x NCOL] = A[M x 128] * W[128 x NCOL], K = 128 fixed.
// One wave computes a 16x32 output strip (two 16x16 tiles sharing the A
// fragment -> A traffic halved, 2 WMMAs per A load). Block = NCOL threads
// (NCOL/32 waves) so one block covers a full 16-row strip across all N.
//
// V_WMMA_F32_16X16X4_F32 fragment layout:
//   A 16x4 : lane m = lane&15; lanes 0-15 hold K={0,1}, lanes 16-31 K={2,3}
//   B 4x16 : lane n = lane&15; same K split across lane halves
//   D 16x16: VGPR r holds row r (lanes 0-15) / row r+8 (lanes 16-31)
//
// Rows >= M: address clamped to row 0; garbage only affects output rows that
// are never stored (lane m's A data only feeds output row m), so no masking.
// ---------------------------------------------------------------------------
template <int NCOL>
__global__ void __launch_bounds__(NCOL)
gemm_wmma_f32(const float* __restrict__ A, const float* __restrict__ W,
              float* __restrict__ D, int M) {
    const int lane = threadIdx.x & 31;
    const int wv   = threadIdx.x >> 5;
    const int half = lane >> 4;      // 0: K pair {0,1}, 1: K pair {2,3}
    const int l16  = lane & 15;
    const int mt   = blockIdx.x;
    const int n0   = (wv * 2) * 16 + l16;   // first tile column
    const int n1   = n0 + 16;               // second tile column
    const int m    = mt * 16 + l16;
    const int kb   = half * 2;

    const float* arow = A + (size_t)(m < M ? m : 0) * 128;

    v8f acc0 = {};
    v8f acc1 = {};
#pragma unroll
    for (int k = 0; k < 128; k += 4) {
        v2f a = *(const v2f*)(arow + k + kb);       // global_load_b64
        v2f b0, b1;
        b0.x = W[(size_t)(k + kb    ) * NCOL + n0];
        b0.y = W[(size_t)(k + kb + 1) * NCOL + n0];
        b1.x = W[(size_t)(k + kb    ) * NCOL + n1];
        b1.y = W[(size_t)(k + kb + 1) * NCOL + n1];
        acc0 = __builtin_amdgcn_wmma_f32_16x16x4_f32(
            false, a, false, b0, (short)0, acc0, false, false);
        acc1 = __builtin_amdgcn_wmma_f32_16x16x4_f32(
            false, a, false, b1, (short)0, acc1, false, false);
    }

    const int rowbase = mt * 16 + half * 8;
#pragma unroll
    for (int r = 0; r < 8; ++r) {
        int row = rowbase + r;
        if (row < M) {
            D[(size_t)row * NCOL + n0] = acc0[r];
            D[(size_t)row * NCOL + n1] = acc1[r];
        }
    }
}

// ---------------------------------------------------------------------------
// aggregation: out[i] = h[i]*dinv[i]^2 (self loop, full overwrite == init)
// ---------------------------------------------------------------------------
template <int F>
__global__ void self_loop_init(const float* __restrict__ h, const float* __restrict__ dinv,
                               float* __restrict__ out, int N) {
    int t = blockIdx.x * blockDim.x + threadIdx.x;
    if (t >= N * F) return;
    int i = t / F;
    float w = dinv[i];
    out[t] = h[t] * (w * w);
}

// one wave per edge: gather h[src] (float4/float2 per lane), scatter-add to out[dst]
template <int F>
__global__ void edge_aggregate(const int* __restrict__ src, const int* __restrict__ dst,
                               const float* __restrict__ dinv,
                               const float* __restrict__ h, float* __restrict__ out, int E) {
    int t    = blockIdx.x * blockDim.x + threadIdx.x;
    int e    = t >> 5;
    int lane = t & 31;
    if (e >= E) return;
    int s = src[e], d = dst[e];
    float norm = dinv[s] * dinv[d];
    constexpr int V = F / 32;           // 4 for F=128, 2 for F=64
    const float* hp = h   + (size_t)s * F + lane * V;
    float*       op = out + (size_t)d * F + lane * V;
    if constexpr (V == 4) {
        float4 v = *(const float4*)hp;  // global_load_b128
        atomicAdd(op + 0, v.x * norm);
        atomicAdd(op + 1, v.y * norm);
        atomicAdd(op + 2, v.z * norm);
        atomicAdd(op + 3, v.w * norm);
    } else {
        float2 v = *(const float2*)hp;
        atomicAdd(op + 0, v.x * norm);
        atomicAdd(op + 1, v.y * norm);
    }
}

template <int F, bool RELU>
__global__ void bias_act(float* __restrict__ x, const float* __restrict__ b, int N) {
    int t = blockIdx.x * blockDim.x + threadIdx.x;
    if (t >= N * F) return;
    float v = x[t] + b[t & (F - 1)];
    if (RELU) v = fmaxf(v, 0.0f);
    x[t] = v;
}

// ---------------------------------------------------------------------------
// global mean pool
// ---------------------------------------------------------------------------
__global__ void pool_count(const int* __restrict__ batch, float* __restrict__ cnt, int N) {
    int i = blockIdx.x * blockDim.x + threadIdx.x;
    if (i < N) atomicAdd(&cnt[batch[i]], 1.0f);
}

__global__ void pool_sum64(const float* __restrict__ x, const int* __restrict__ batch,
                           float* __restrict__ gsum, int N) {
    int t    = blockIdx.x * blockDim.x + threadIdx.x;
    int i    = t >> 5;
    int lane = t & 31;
    if (i >= N) return;
    int g = batch[i];
    float2 v = *(const float2*)(x + (size_t)i * 64 + lane * 2);
    atomicAdd(&gsum[(size_t)g * 64 + lane * 2 + 0], v.x);
    atomicAdd(&gsum[(size_t)g * 64 + lane * 2 + 1], v.y);
}

__global__ void pool_div(const float* __restrict__ gsum, const float* __restrict__ cnt,
                         float* __restrict__ out, int G) {
    int t = blockIdx.x * blockDim.x + threadIdx.x;
    if (t >= G * 64) return;
    out[t] = gsum[t] / fmaxf(cnt[t >> 6], 1.0f);
}

// ---------------------------------------------------------------------------
static inline int cdiv_i(long long a, long long b) { return (int)((a + b - 1) / b); }

extern "C" void kernel_launch(void* const* d_in, const int* in_sizes, int n_in,
                              void* d_out, int out_size, void* d_ws, size_t ws_size,
                              hipStream_t stream) {
    const float* x     = (const float*)d_in[0];
    const int*   ei    = (const int*)d_in[1];   // [2, E] row-major
    const int*   batch = (const int*)d_in[2];
    const float* W1    = (const float*)d_in[3];
    const float* b1    = (const float*)d_in[4];
    const float* W2    = (const float*)d_in[5];
    const float* b2    = (const float*)d_in[6];
    float*       out   = (float*)d_out;

    const int N = in_sizes[0] / 128;
    const int E = in_sizes[1] / 2;
    const int G = out_size / 64;
    const int* src = ei;
    const int* dst = ei + E;

    // workspace layout (all accumulated buffers fully re-initialized each call)
    float* bufA = (float*)d_ws;              // N*128 : h1, later h2 | agg2
    float* bufB = bufA + (size_t)N * 128;    // N*128 : agg1
    float* dinv = bufB + (size_t)N * 128;    // N
    float* gsum = dinv + N;                  // G*64
    float* gcnt = gsum + (size_t)G * 64;     // G   (contiguous after gsum)

    // degree -> dinv
    fill_f32<<<cdiv_i(N, 256), 256, 0, stream>>>(dinv, 1.0f, N);  // self loop
    deg_accum<<<cdiv_i(E, 256), 256, 0, stream>>>(dst, dinv, E);
    deg_to_dinv<<<cdiv_i(N, 256), 256, 0, stream>>>(dinv, N);

    // ---- layer 1: h1 = x @ W1 ; agg1 = sym-norm aggregate ; relu(+b1)
    gemm_wmma_f32<128><<<cdiv_i(N, 16), 128, 0, stream>>>(x, W1, bufA, N);
    self_loop_init<128><<<cdiv_i((long long)N * 128, 256), 256, 0, stream>>>(bufA, dinv, bufB, N);
    edge_aggregate<128><<<cdiv_i((long long)E * 32, 256), 256, 0, stream>>>(src, dst, dinv, bufA, bufB, E);
    bias_act<128, true><<<cdiv_i((long long)N * 128, 256), 256, 0, stream>>>(bufB, b1, N);

    // ---- layer 2: h2 = agg1 @ W2 ; agg2 ; +b2   (h2/agg2 overlay dead h1)
    float* h2   = bufA;
    float* agg2 = bufA + (size_t)N * 64;
    gemm_wmma_f32<64><<<cdiv_i(N, 16), 64, 0, stream>>>(bufB, W2, h2, N);
    self_loop_init<64><<<cdiv_i((long long)N * 64, 256), 256, 0, stream>>>(h2, dinv, agg2, N);
    edge_aggregate<64><<<cdiv_i((long long)E * 32, 256), 256, 0, stream>>>(src, dst, dinv, h2, agg2, E);
    bias_act<64, false><<<cdiv_i((long long)N * 64, 256), 256, 0, stream>>>(agg2, b2, N);

    // ---- global mean pool
    fill_f32<<<cdiv_i((long long)G * 64 + G, 256), 256, 0, stream>>>(gsum, 0.0f, G * 64 + G);
    pool_count<<<cdiv_i(N, 256), 256, 0, stream>>>(batch, gcnt, N);
    pool_sum64<<<cdiv_i((long long)N * 32, 256), 256, 0, stream>>>(agg2, batch, gsum, N);
    pool_div<<<cdiv_i((long long)G * 64, 256), 256, 0, stream>>>(gsum, gcnt, out, G);
}